// GraphSageLinkPredictor_84335977824484
// MI455X (gfx1250) — compile-verified
//
#include <hip/hip_runtime.h>

typedef __attribute__((ext_vector_type(2))) float v2f;
typedef __attribute__((ext_vector_type(8))) float v8f;

#define NNODES 100000
#define DIM 128
#define LDSPITCH 132   // break bank conflicts on column reads (132 % 64 == 4)

// ---------------------------------------------------------------- zero buffer
__global__ void k_zero(float* __restrict__ p, long long n) {
    long long i = (long long)blockIdx.x * blockDim.x + threadIdx.x;
    if (i < n) p[i] = 0.0f;
}

// ---------------------------------------------------------------- degree: atomicAdd 1 per edge, then invert
__global__ void k_deg(const long long* __restrict__ dst, float* __restrict__ deg, int E) {
    int e = blockIdx.x * blockDim.x + threadIdx.x;
    if (e < E) atomicAdd(&deg[(int)dst[e]], 1.0f);
}

__global__ void k_invdeg(float* __restrict__ deg, int n) {
    int i = blockIdx.x * blockDim.x + threadIdx.x;
    if (i < n) deg[i] = 1.0f / fmaxf(deg[i], 1.0f);
}

// ---------------------------------------------------------------- scatter: agg[dst] += x[src]
// one wave (32 lanes) per edge, each lane moves 4 consecutive floats
__global__ void k_scatter(const float* __restrict__ x,
                          const long long* __restrict__ src,
                          const long long* __restrict__ dst,
                          float* __restrict__ agg, int E) {
    int t    = blockIdx.x * blockDim.x + threadIdx.x;
    int e    = t >> 5;
    int lane = t & 31;
    if (e >= E) return;
    int s = (int)src[e];
    int d = (int)dst[e];
    const float4 v = ((const float4*)(x + (size_t)s * DIM))[lane];
    float* ap = agg + (size_t)d * DIM + lane * 4;
    atomicAdd(ap + 0, v.x);
    atomicAdd(ap + 1, v.y);
    atomicAdd(ap + 2, v.z);
    atomicAdd(ap + 3, v.w);
}

// ---------------------------------------------------------------- fused SAGE layer GEMM
// out[16 rows x 128 cols per block] = act( (agg*invdeg) @ Wn + x @ Ws + b (+ x residual) )
// block = 256 threads = 8 waves; wave w computes column tile n0 = 16*w.
// K-loop uses V_WMMA_F32_16X16X4_F32 (fp32 exact, matches reference dtype).
template <bool RELU, bool RESIDUAL>
__global__ void k_sage_gemm(const float* __restrict__ agg,
                            const float* __restrict__ x,
                            const float* __restrict__ invdeg,
                            const float* __restrict__ Wn,
                            const float* __restrict__ Ws,
                            const float* __restrict__ bias,
                            float* __restrict__ out) {
    __shared__ float sAgg[16 * LDSPITCH];
    __shared__ float sX[16 * LDSPITCH];

    const int tid  = threadIdx.x;
    const int row0 = blockIdx.x * 16;

    // cooperative, coalesced stage of the 16x128 A-tiles into LDS
    for (int i = tid; i < 16 * DIM; i += 256) {
        int r = i >> 7;        // 0..15
        int c = i & (DIM - 1); // 0..127
        size_t g = (size_t)(row0 + r) * DIM + c;
        sAgg[r * LDSPITCH + c] = agg[g] * invdeg[row0 + r];
        sX[r * LDSPITCH + c]   = x[g];
    }
    __syncthreads();

    const int wave = tid >> 5;
    const int lane = tid & 31;
    const int n0   = wave * 16;
    const int m    = lane & 15;          // A: M index / B,C: N index
    const int koff = (lane >> 4) << 1;   // lanes 0-15 -> K {0,1}; 16-31 -> K {2,3}

    v8f c = {};
    for (int k = 0; k < DIM; k += 4) {
        // A fragment from scaled-agg tile, B fragment from Wn
        v2f a0, b0;
        a0.x = sAgg[m * LDSPITCH + k + koff];
        a0.y = sAgg[m * LDSPITCH + k + koff + 1];
        b0.x = Wn[(size_t)(k + koff) * DIM + n0 + m];
        b0.y = Wn[(size_t)(k + koff + 1) * DIM + n0 + m];
        c = __builtin_amdgcn_wmma_f32_16x16x4_f32(false, a0, false, b0,
                                                  (short)0, c, false, false);
        // A fragment from x tile, B fragment from Ws
        v2f a1, b1;
        a1.x = sX[m * LDSPITCH + k + koff];
        a1.y = sX[m * LDSPITCH + k + koff + 1];
        b1.x = Ws[(size_t)(k + koff) * DIM + n0 + m];
        b1.y = Ws[(size_t)(k + koff + 1) * DIM + n0 + m];
        c = __builtin_amdgcn_wmma_f32_16x16x4_f32(false, a1, false, b1,
                                                  (short)0, c, false, false);
    }

    // epilogue: C/D layout -> vgpr r: M = r + 8*(lane>>4), N = lane&15
    const int   col = n0 + m;
    const float bv  = bias[col];
    const int   mhi = (lane >> 4) << 3;
#pragma unroll
    for (int r = 0; r < 8; ++r) {
        int    row = row0 + r + mhi;
        size_t g   = (size_t)row * DIM + col;
        float  v   = c[r] + bv;
        if (RESIDUAL) v += x[g];
        if (RELU) v = fmaxf(v, 0.0f);
        out[g] = v;
    }
}

// ---------------------------------------------------------------- link decoder: logits = sum(h[s]*h[d])
// one wave per label edge; each lane owns 4 floats (float4), shfl_xor reduce
__global__ void k_decode(const float* __restrict__ h,
                         const long long* __restrict__ ls,
                         const long long* __restrict__ ld,
                         float* __restrict__ out, int EL) {
    int t    = blockIdx.x * blockDim.x + threadIdx.x;
    int e    = t >> 5;
    int lane = t & 31;
    if (e >= EL) return;
    int s = (int)ls[e];
    int d = (int)ld[e];
    const float4 a = ((const float4*)(h + (size_t)s * DIM))[lane];
    const float4 b = ((const float4*)(h + (size_t)d * DIM))[lane];
    float acc = a.x * b.x + a.y * b.y + a.z * b.z + a.w * b.w;
#pragma unroll
    for (int o = 16; o > 0; o >>= 1) acc += __shfl_xor(acc, o, 32);
    if (lane == 0) out[e] = acc;
}

// ---------------------------------------------------------------- host orchestration
extern "C" void kernel_launch(void* const* d_in, const int* in_sizes, int n_in,
                              void* d_out, int out_size, void* d_ws, size_t ws_size,
                              hipStream_t stream) {
    const float*     xin  = (const float*)d_in[0];      // [N, 128]
    const long long* ei   = (const long long*)d_in[1];  // [2, E] int64
    const long long* eli  = (const long long*)d_in[2];  // [2, EL] int64
    const float*     wn   = (const float*)d_in[3];      // [3, 128, 128]
    const float*     ws   = (const float*)d_in[4];      // [3, 128, 128]
    const float*     bs   = (const float*)d_in[5];      // [3, 128]
    float*           outp = (float*)d_out;

    const int E  = in_sizes[1] / 2;
    const int EL = in_sizes[2] / 2;
    const long long ND = (long long)NNODES * DIM;

    const long long* esrc = ei;
    const long long* edst = ei + E;
    const long long* lsrc = eli;
    const long long* ldst = eli + EL;

    float* bufA = (float*)d_ws;        // h after layers 0 / 2
    float* bufB = bufA + ND;           // h after layer 1
    float* agg  = bufB + ND;           // neighbor accumulator
    float* ideg = agg + ND;            // degree -> inverse degree
    (void)ws_size; (void)n_in; (void)out_size;

    const int ZB = (int)((ND + 255) / 256);
    const int SB = (int)(((long long)E * 32 + 255) / 256);
    const int GB = NNODES / 16;        // 6250 row tiles

    // inverse degree (shared by all layers)
    k_zero<<<(NNODES + 255) / 256, 256, 0, stream>>>(ideg, NNODES);
    k_deg<<<(E + 255) / 256, 256, 0, stream>>>(edst, ideg, E);
    k_invdeg<<<(NNODES + 255) / 256, 256, 0, stream>>>(ideg, NNODES);

    // layer 0: relu(conv(x)) -> bufA
    k_zero<<<ZB, 256, 0, stream>>>(agg, ND);
    k_scatter<<<SB, 256, 0, stream>>>(xin, esrc, edst, agg, E);
    k_sage_gemm<true, false><<<GB, 256, 0, stream>>>(agg, xin, ideg,
                                                     wn, ws, bs, bufA);

    // layer 1: relu(conv(bufA) + bufA) -> bufB
    k_zero<<<ZB, 256, 0, stream>>>(agg, ND);
    k_scatter<<<SB, 256, 0, stream>>>(bufA, esrc, edst, agg, E);
    k_sage_gemm<true, true><<<GB, 256, 0, stream>>>(agg, bufA, ideg,
                                                    wn + 128 * 128, ws + 128 * 128,
                                                    bs + 128, bufB);

    // layer 2: conv(bufB) + bufB -> bufA
    k_zero<<<ZB, 256, 0, stream>>>(agg, ND);
    k_scatter<<<SB, 256, 0, stream>>>(bufB, esrc, edst, agg, E);
    k_sage_gemm<false, true><<<GB, 256, 0, stream>>>(agg, bufB, ideg,
                                                     wn + 2 * 128 * 128, ws + 2 * 128 * 128,
                                                     bs + 2 * 128, bufA);

    // dot-product link decoder
    k_decode<<<(int)(((long long)EL * 32 + 255) / 256), 256, 0, stream>>>(
        bufA, lsrc, ldst, outp, EL);
}